// AbsolutePositionalEncoder_73993696575862
// MI455X (gfx1250) — compile-verified
//
#include <hip/hip_runtime.h>
#include <hip/hip_bf16.h>
#include <stdint.h>

// ---------------------------------------------------------------------------
// AbsolutePositionalEncoder: out[b,t,:] = W[idx[b,t],:] + bias
//   idx : [8192,64] int32       W : [365,128] f32      bias : [128] f32
//   out : [8192,64,128] f32  (256 MB streaming writes -> HBM-bound, ~11us)
//
// Strategy (MI455X / gfx1250):
//   * W (186,880 B) fits in the 320 KB WGP LDS -> stage it once per block
//     with CDNA5 async global->LDS B128 copies (ASYNCcnt), then gather rows
//     from LDS with ds_load_b128 (deterministic latency, no L1 thrash from
//     random 512 B row reads).
//   * One wave per row: lane l handles float4 chunk l of the 128-float row.
//   * Non-temporal global_store_b128 for the 256 MB write-once output.
// ---------------------------------------------------------------------------

typedef float v4f __attribute__((ext_vector_type(4)));

#define NUM_DAYS 365
#define D_MODEL  128
#define BT_ROWS  (8192 * 64)          // 524288 rows
#define BLOCK    1024                 // 32 waves: one block fills a WGP
#define GRID     512                  // 1024 contiguous rows per block
#define W_V4     ((NUM_DAYS * D_MODEL) / 4)   // 11680 float4 chunks (186,880 B)

static_assert(BT_ROWS % GRID == 0, "rows divide evenly");
static_assert((BT_ROWS / GRID) % 32 == 0, "rows per block divide by waves");

__global__ __launch_bounds__(BLOCK)
void pe_gather_kernel(const int* __restrict__ idx,
                      const float* __restrict__ W,
                      const float* __restrict__ bias,
                      float* __restrict__ out)
{
    __shared__ v4f ldsW[W_V4];        // 186,880 bytes of the 320 KB WGP LDS

    // ---- Stage W into LDS with CDNA5 async copies (tracked on ASYNCcnt) ----
    {
        const v4f* Wv = (const v4f*)W;
        // Generic pointers into the LDS aperture carry the LDS byte offset in
        // their low 32 bits (ISA 10.2 aperture mapping), so this is the VDST
        // LDS address the instruction wants.
        unsigned lds_base = (unsigned)(uintptr_t)&ldsW[0];
        for (int i = threadIdx.x; i < W_V4; i += BLOCK) {
            unsigned long long ga = (unsigned long long)(uintptr_t)(Wv + i);
            unsigned la = lds_base + (unsigned)i * 16u;
            asm volatile("global_load_async_to_lds_b128 %0, %1, off"
                         :: "v"(la), "v"(ga) : "memory");
        }
        asm volatile("s_wait_asynccnt 0" ::: "memory");
        __syncthreads();              // cross-wave visibility of staged W
    }

    const int lane = threadIdx.x & 31;
    const int wave = threadIdx.x >> 5;

    // Bias chunk for this lane, loaded once (hot in cache).
    const v4f bv = ((const v4f*)bias)[lane];

    constexpr int ROWS_PER_BLOCK = BT_ROWS / GRID;   // 1024
    const int row0 = blockIdx.x * ROWS_PER_BLOCK + wave;

    v4f* __restrict__ out4 = (v4f*)out;

    #pragma unroll 8
    for (int r = 0; r < ROWS_PER_BLOCK; r += 32) {
        const int row = row0 + r;
        const int day = idx[row];                    // wave-uniform 4B load
        // LDS gather: lane l reads bytes [day*512 + l*16, +16)
        v4f v = ldsW[day * (D_MODEL / 4) + lane] + bv;
        // Streaming write-once output: non-temporal B128 store
        __builtin_nontemporal_store(v, &out4[(size_t)row * (D_MODEL / 4) + lane]);
    }
}

extern "C" void kernel_launch(void* const* d_in, const int* in_sizes, int n_in,
                              void* d_out, int out_size, void* d_ws, size_t ws_size,
                              hipStream_t stream) {
    const int*   idx  = (const int*)d_in[0];    // [8192,64] int32
    const float* W    = (const float*)d_in[1];  // [365,128] f32
    const float* bias = (const float*)d_in[2];  // [128] f32
    float*       out  = (float*)d_out;          // [8192,64,128] f32

    (void)in_sizes; (void)n_in; (void)out_size; (void)d_ws; (void)ws_size;

    pe_gather_kernel<<<GRID, BLOCK, 0, stream>>>(idx, W, bias, out);
}